// MemAggregator_34067680592246
// MI455X (gfx1250) — compile-verified
//
#include <hip/hip_runtime.h>

#define NS 8192
#define DD 16
#define CC 8
#define RR 64
#define ROWS 152          // CC + DD*(CC+1)
#define NEG_INF (-3.402823466e38f)

// ---- CDNA5 async global->LDS staging (ASYNCcnt path), guarded ------------
#if defined(__gfx1250__) && __has_builtin(__builtin_amdgcn_global_load_async_to_lds_b128)
#define ASYNC_LDS 1
typedef int v4i_t __attribute__((ext_vector_type(4)));
typedef __attribute__((address_space(1))) v4i_t gas_v4i;   // global
typedef __attribute__((address_space(3))) v4i_t las_v4i;   // LDS
#else
#define ASYNC_LDS 0
#endif

__device__ __forceinline__ void async_copy16(const float* gsrc, float* lds_dst) {
#if ASYNC_LDS
    __builtin_amdgcn_global_load_async_to_lds_b128(
        (gas_v4i*)gsrc, (las_v4i*)lds_dst, 0, 0);
#else
    *(float4*)lds_dst = *(const float4*)gsrc;
#endif
}

__device__ __forceinline__ void wait_async_lds() {
#if ASYNC_LDS
#if __has_builtin(__builtin_amdgcn_s_wait_asynccnt)
    __builtin_amdgcn_s_wait_asynccnt(0);
#else
    asm volatile("s_wait_asynccnt 0" ::: "memory");
#endif
#endif
}

// ---- Kernel 1: new_emb + per-row scores + top-8 --------------------------
__global__ __launch_bounds__(256) void score_topk_kernel(
    const float* __restrict__ curr_emb, const float* __restrict__ alpha,
    const float* __restrict__ msg, const float* __restrict__ curr_node_mem,
    const float* __restrict__ curr_rel_mem, const float* __restrict__ nei_node_mem,
    const float* __restrict__ nei_rel_mem, const float* __restrict__ head_rel_emb,
    const float* __restrict__ head_emb, float* __restrict__ out_emb,
    int* __restrict__ topk)
{
    __shared__ __align__(16) float s_msg[DD * RR];
    __shared__ __align__(16) float s_hre[DD * RR];
    __shared__ __align__(16) float s_he [DD * RR];
    __shared__ __align__(16) float s_emb[RR];
    __shared__ float s_scores[160];
    __shared__ int   s_top[CC];

    const int n   = blockIdx.x;
    const int tid = threadIdx.x;
    const size_t nDR = (size_t)n * (DD * RR);   // n*1024

    // Stage the three per-n (16x64 f32) tiles into LDS asynchronously.
    {
        const int off = tid * 4;                // 256 threads * 4 floats = 1024
        async_copy16(msg          + nDR + off, s_msg + off);
        async_copy16(head_rel_emb + nDR + off, s_hre + off);
        async_copy16(head_emb     + nDR + off, s_he  + off);
        wait_async_lds();
    }
    if (tid >= ROWS && tid < 160) s_scores[tid] = NEG_INF;   // pad
    __syncthreads();

    // Phase A: new_emb[r] = ce[r] + sum_d alpha[d]*msg[d][r]
    if (tid < RR) {
        float acc = curr_emb[nDR + tid];        // curr_emb[n][0][r]
        #pragma unroll
        for (int d = 0; d < DD; ++d)
            acc += alpha[(size_t)n * DD + d] * s_msg[d * RR + tid];
        s_emb[tid] = acc;
        out_emb[(size_t)n * RR + tid] = acc;
    }
    __syncthreads();

    // Phase B: one thread per candidate row; score = ||node+rel-e||^2
    if (tid < ROWS) {
        const float4* e4 = (const float4*)s_emb;
        float score = 0.f;
        if (tid < CC) {
            const float4* np = (const float4*)(curr_node_mem + ((size_t)n * (DD * CC) + tid) * RR);
            const float4* rp = (const float4*)(curr_rel_mem  + ((size_t)n * (DD * CC) + tid) * RR);
            #pragma unroll
            for (int i = 0; i < RR / 4; ++i) {
                float4 a = np[i], b = rp[i], e = e4[i];
                float dx = a.x + b.x - e.x, dy = a.y + b.y - e.y;
                float dz = a.z + b.z - e.z, dw = a.w + b.w - e.w;
                score += dx * dx + dy * dy + dz * dz + dw * dw;
            }
        } else {
            const int m = tid - CC;
            const int d = m / 9;
            const int j = m - d * 9;
            const float4* h4 = (const float4*)(s_hre + d * RR);
            if (j < CC) {
                const size_t base = (((size_t)n * DD + d) * CC + j) * RR;
                const float4* np = (const float4*)(nei_node_mem + base);
                const float4* rp = (const float4*)(nei_rel_mem + base);
                #pragma unroll
                for (int i = 0; i < RR / 4; ++i) {
                    float4 a = np[i], b = rp[i], h = h4[i], e = e4[i];
                    float dx = a.x + b.x + h.x - e.x, dy = a.y + b.y + h.y - e.y;
                    float dz = a.z + b.z + h.z - e.z, dw = a.w + b.w + h.w - e.w;
                    score += dx * dx + dy * dy + dz * dz + dw * dw;
                }
            } else {
                const float4* p4 = (const float4*)(s_he + d * RR);
                #pragma unroll
                for (int i = 0; i < RR / 4; ++i) {
                    float4 a = p4[i], h = h4[i], e = e4[i];
                    float dx = a.x + h.x - e.x, dy = a.y + h.y - e.y;
                    float dz = a.z + h.z - e.z, dw = a.w + h.w - e.w;
                    score += dx * dx + dy * dy + dz * dz + dw * dw;
                }
            }
        }
        s_scores[tid] = score;
    }
    __syncthreads();

    // Phase C: 8 x wave32 argmax (strict '>' + smallest-index tie-break
    // replicates stable argsort of -score).
    for (int it = 0; it < CC; ++it) {
        if (tid < 32) {
            float best = NEG_INF;
            int   bi   = 0x7fffffff;
            for (int k = tid; k < 160; k += 32) {
                float v = s_scores[k];
                if (v > best) { best = v; bi = k; }
            }
            #pragma unroll
            for (int off = 16; off > 0; off >>= 1) {
                float ob = __shfl_xor(best, off, 32);
                int   oi = __shfl_xor(bi,   off, 32);
                if (ob > best || (ob == best && oi < bi)) { best = ob; bi = oi; }
            }
            if (tid == 0) { s_top[it] = bi; s_scores[bi] = NEG_INF; }
        }
        __syncthreads();
    }
    if (tid < CC) topk[(size_t)n * CC + tid] = s_top[tid];
}

// ---- Kernel 2: replicate the cross-sample gather exactly -----------------
// j = n*8 + t ; source sample n' = j/152, source row k' = j%152
__global__ __launch_bounds__(256) void gather_kernel(
    const float* __restrict__ curr_node_mem, const float* __restrict__ curr_rel_mem,
    const float* __restrict__ nei_node_mem, const float* __restrict__ nei_rel_mem,
    const float* __restrict__ head_rel_emb, const float* __restrict__ head_emb,
    const int* __restrict__ topk,
    float* __restrict__ out_node, float* __restrict__ out_rel)
{
    const int gid = blockIdx.x * 256 + threadIdx.x;   // N*C*R threads total
    const int r   = gid & (RR - 1);
    const int row = gid >> 6;                         // n*8 + i  (0..65535)
    const int n   = row >> 3;

    const int t  = topk[row];
    const int j  = n * CC + t;
    const int np = j / ROWS;
    const int kp = j - np * ROWS;

    float nodev, relv;
    if (kp < CC) {
        const size_t base = ((size_t)np * (DD * CC) + kp) * RR + r;  // [np][0][kp][r]
        nodev = curr_node_mem[base];
        relv  = curr_rel_mem[base];
    } else {
        const int m  = kp - CC;
        const int d  = m / 9;
        const int jj = m - d * 9;
        const size_t hb = ((size_t)np * DD + d) * RR + r;
        const float hre = head_rel_emb[hb];
        if (jj < CC) {
            const size_t base = (((size_t)np * DD + d) * CC + jj) * RR + r;
            nodev = nei_node_mem[base];
            relv  = nei_rel_mem[base] + hre;
        } else {
            nodev = head_emb[hb];
            relv  = hre;
        }
    }
    out_node[(size_t)row * RR + r] = nodev;
    out_rel [(size_t)row * RR + r] = relv;
}

extern "C" void kernel_launch(void* const* d_in, const int* in_sizes, int n_in,
                              void* d_out, int out_size, void* d_ws, size_t ws_size,
                              hipStream_t stream) {
    const float* curr_emb      = (const float*)d_in[0];
    const float* alpha         = (const float*)d_in[1];
    const float* msg           = (const float*)d_in[2];
    const float* curr_node_mem = (const float*)d_in[3];
    const float* curr_rel_mem  = (const float*)d_in[4];
    const float* nei_node_mem  = (const float*)d_in[5];
    const float* nei_rel_mem   = (const float*)d_in[6];
    const float* head_rel_emb  = (const float*)d_in[7];
    const float* head_emb      = (const float*)d_in[8];

    float* out_emb  = (float*)d_out;                            // (N,64)
    float* out_node = out_emb + (size_t)NS * RR;                // (N,8,64)
    float* out_rel  = out_node + (size_t)NS * CC * RR;          // (N,8,64)
    int*   topk     = (int*)d_ws;                               // N*8 ints

    score_topk_kernel<<<NS, 256, 0, stream>>>(
        curr_emb, alpha, msg, curr_node_mem, curr_rel_mem,
        nei_node_mem, nei_rel_mem, head_rel_emb, head_emb,
        out_emb, topk);

    const int total = NS * CC * RR;                             // 4,194,304
    gather_kernel<<<total / 256, 256, 0, stream>>>(
        curr_node_mem, curr_rel_mem, nei_node_mem, nei_rel_mem,
        head_rel_emb, head_emb, topk, out_node, out_rel);
}